// AdvectionDiffusion_43344809951329
// MI455X (gfx1250) — compile-verified
//
#include <hip/hip_runtime.h>
#include <math.h>
#include <stdint.h>

// ---------------------------------------------------------------------------
// Advection-diffusion time march, 512x512, 255 sequential steps.
//
// MI455X roofline: 255 * 1 MB = 267 MB of mandatory HBM writes @ 23.3 TB/s
// ~= 11.5 us; per-step state (1 MB) is L2-resident (192 MB L2) so reads are
// ~free; total compute ~1.3 GFLOP fp32 => negligible. A 255-kernel chain
// would be dominated by ~1-2 us/launch dispatch latency (~300-500 us), so we
// run ONE persistent kernel (64 blocks -- trivially co-resident) that steps
// the PDE 255 times with a device-wide atomic barrier between steps.
//
// CDNA5 paths used: GLOBAL_LOAD_ASYNC_TO_LDS_B128 (ASYNCcnt) for halo-tile
// and coefficient staging, s_wait_asynccnt, s_sleep spin barrier, b128 output
// stores. WMMA is inapplicable (5-point stencil, no matrix op).
// ---------------------------------------------------------------------------

#define GN 512
#define GNG 4
#define STEPS 255
#define PBLOCKS 64          // persistent blocks (8 rows each); conservative
#define PROWS (GN / PBLOCKS) // 8 output rows per block
#define HROWS (PROWS + 2)    // 10 staged rows (with halo)

static constexpr float LBOX = 10.0f;
static constexpr float DTC  = 5e-05f;

typedef int v4i __attribute__((ext_vector_type(4)));

__device__ __forceinline__ int iclamp(int v, int lo, int hi) {
  return v < lo ? lo : (v > hi ? hi : v);
}

// ---- CDNA5 async global->LDS copy (ASYNCcnt) ------------------------------
__device__ __forceinline__ void async_copy_b128(const float* g, float* l) {
#if defined(__HIP_DEVICE_COMPILE__)
#if __has_builtin(__builtin_amdgcn_global_load_async_to_lds_b128)
  __builtin_amdgcn_global_load_async_to_lds_b128(
      (v4i*)(uintptr_t)g, (v4i*)(uintptr_t)l, /*offset=*/0, /*cpol=*/0);
#else
  asm volatile("global_load_async_to_lds_b128 %0, %1, off"
               :: "v"((unsigned int)(uintptr_t)l),
                  "v"((unsigned long long)(uintptr_t)g)
               : "memory");
#endif
#else
  (void)g; (void)l;
#endif
}

__device__ __forceinline__ void async_wait_all() {
#if defined(__HIP_DEVICE_COMPILE__)
#if __has_builtin(__builtin_amdgcn_s_wait_asynccnt)
  __builtin_amdgcn_s_wait_asynccnt(0);
#else
  asm volatile("s_wait_asynccnt 0" ::: "memory");
#endif
#endif
}

// ---- precompute Gaussian coefficient maps into workspace ------------------
__global__ __launch_bounds__(256)
void gauss_maps_kernel(const float* __restrict__ alpha,
                       const float* __restrict__ kappa,
                       float* __restrict__ amap,
                       float* __restrict__ kmap) {
  int idx = blockIdx.x * 256 + threadIdx.x;
  if (idx >= GN * GN) return;
  int i = idx / GN;
  int j = idx % GN;
  const float h = LBOX / (float)(GN - 1);   // linspace(-5, 5, 512)
  float x = -5.0f + h * (float)i;
  float y = -5.0f + h * (float)j;
  float a = 0.1f;
  float k = 0.1f;
#pragma unroll
  for (int g = 0; g < GNG; ++g) {
    float dxa = x - alpha[GNG + g];
    float dya = y - alpha[2 * GNG + g];
    a += alpha[g] * expf(-(dxa * dxa + dya * dya));
    float dxk = x - kappa[GNG + g];
    float dyk = y - kappa[2 * GNG + g];
    k += kappa[g] * expf(-(dxk * dxk + dyk * dyk));
  }
  amap[idx] = a;
  kmap[idx] = k;
}

// ---- reset device-wide barrier counter (poisoned workspace) ---------------
__global__ void init_counter_kernel(unsigned* __restrict__ c) {
  if (threadIdx.x == 0 && blockIdx.x == 0) *c = 0u;
}

// ---- persistent time-march kernel -----------------------------------------
__global__ __launch_bounds__(256)
void march_kernel(const float* __restrict__ u0,
                  const float* __restrict__ amap,
                  const float* __restrict__ kmap,
                  float* __restrict__ out,
                  unsigned* __restrict__ bar) {
  __shared__ float tileT[HROWS][GN];   // 10 * 2 KB = 20 KB
  __shared__ float tileA[PROWS][GN];   //  8 * 2 KB = 16 KB
  __shared__ float tileK[PROWS][GN];   //  8 * 2 KB = 16 KB

  const int t  = threadIdx.x;
  const int r0 = blockIdx.x * PROWS;   // first output row of this block

  // Stage step-invariant coefficient rows once (row-clamped to interior).
#pragma unroll
  for (int f = t; f < PROWS * (GN / 4); f += 256) {
    int lr = f / (GN / 4);
    int c4 = (f % (GN / 4)) * 4;
    int ci = iclamp(r0 + lr, 1, GN - 2);
    async_copy_b128(amap + (size_t)ci * GN + c4, &tileA[lr][c4]);
    async_copy_b128(kmap + (size_t)ci * GN + c4, &tileK[lr][c4]);
  }

  const float h       = LBOX / (float)(GN - 1);
  const float inv_dx2 = 1.0f / (h * h);
  const float inv_2dx = 1.0f / (2.0f * h);

  for (int s = 0; s < STEPS; ++s) {
    const float* prev = (s == 0) ? u0 : out + (size_t)(s - 1) * GN * GN;
    float*       next = out + (size_t)s * GN * GN;

    // Stage rows (r0-1 .. r0+8), clamped, into LDS: 10*128 b128 chunks.
#pragma unroll
    for (int f = t; f < HROWS * (GN / 4); f += 256) {
      int sr = f / (GN / 4);
      int c4 = (f % (GN / 4)) * 4;
      int gr = iclamp(r0 - 1 + sr, 0, GN - 1);
      async_copy_b128(prev + (size_t)gr * GN + c4, &tileT[sr][c4]);
    }
    async_wait_all();
    __syncthreads();

    // 8 rows * 128 float4 = 1024 chunks; 256 threads -> 4 each.
#pragma unroll
    for (int q = 0; q < 4; ++q) {
      int id = t + q * 256;              // 0..1023
      int lr = id / (GN / 4);            // local output row 0..7
      int c4 = (id % (GN / 4)) * 4;      // first column of float4
      int gi = r0 + lr;                  // global output row
      int ci = iclamp(gi, 1, GN - 2);    // edge-pad clamp (rows)
      int sr = ci - (r0 - 1);            // LDS row of stencil center

      float4 outv;
      float* po = &outv.x;
#pragma unroll
      for (int e = 0; e < 4; ++e) {
        int j  = c4 + e;
        int cj = iclamp(j, 1, GN - 2);   // edge-pad clamp (cols)
        float Tc = tileT[sr][cj];
        float Tm = tileT[sr - 1][cj];    // row i-1
        float Tp = tileT[sr + 1][cj];    // row i+1
        float Tl = tileT[sr][cj - 1];    // col j-1
        float Tr = tileT[sr][cj + 1];    // col j+1
        float Txx = (Tm - 2.0f * Tc + Tp) * inv_dx2;
        float Tyy = (Tl - 2.0f * Tc + Tr) * inv_dx2;
        float Tx  = (Tp - Tm) * inv_2dx;
        float Ty  = (Tr - Tl) * inv_2dx;
        float a = tileA[lr][cj];
        float k = tileK[lr][cj];
        po[e] = Tc - DTC * k * (Tc * Tx + Tc * Ty) + DTC * a * (Txx + Tyy);
      }
      *(float4*)(next + (size_t)gi * GN + c4) = outv;
    }

    // ---- device-wide barrier: release -> arrive -> spin -> acquire -------
    __threadfence();        // each wave drains/fences its stores (release)
    __syncthreads();        // whole block done before leader arrives
    if (t == 0) {
      atomicAdd(bar, 1u);   // monotonic count: no reset race
      unsigned target = (unsigned)(s + 1) * PBLOCKS;
      while (__hip_atomic_load(bar, __ATOMIC_RELAXED,
                               __HIP_MEMORY_SCOPE_AGENT) < target) {
        __builtin_amdgcn_s_sleep(2);
      }
    }
    __syncthreads();        // also protects tileT reuse next iteration
    __threadfence();        // acquire side before reading remote rows
  }
}

// ---------------------------------------------------------------------------
extern "C" void kernel_launch(void* const* d_in, const int* in_sizes, int n_in,
                              void* d_out, int out_size, void* d_ws, size_t ws_size,
                              hipStream_t stream) {
  const float* u0    = (const float*)d_in[0];
  const float* alpha = (const float*)d_in[1];
  const float* kappa = (const float*)d_in[2];
  float* out  = (float*)d_out;
  float* amap = (float*)d_ws;                 // 512*512 floats
  float* kmap = amap + (size_t)GN * GN;       // 512*512 floats
  unsigned* bar = (unsigned*)(kmap + (size_t)GN * GN);  // barrier counter

  gauss_maps_kernel<<<(GN * GN + 255) / 256, 256, 0, stream>>>(alpha, kappa,
                                                               amap, kmap);
  init_counter_kernel<<<1, 64, 0, stream>>>(bar);
  march_kernel<<<PBLOCKS, 256, 0, stream>>>(u0, amap, kmap, out, bar);
}